// MyNet2_16372415333131
// MI455X (gfx1250) — compile-verified
//
#include <hip/hip_runtime.h>

typedef float v2f __attribute__((ext_vector_type(2)));
typedef float v8f __attribute__((ext_vector_type(8)));
typedef float v4f __attribute__((ext_vector_type(4)));
typedef int   v4i __attribute__((ext_vector_type(4)));

#define NPG 38   // nodes per graph (K dim of head layer 1)

__device__ __forceinline__ float leaky(float v) {
    return (v >= 0.0f) ? v : 0.01f * v;
}

// ---------------------------------------------------------------------------
// Kernel 1: nodes[i] = x[i]*w_root + b_conv   (base term; agg is added on top)
// ---------------------------------------------------------------------------
__global__ __launch_bounds__(256) void init_nodes_kernel(
    const float* __restrict__ x, const float* __restrict__ w_root,
    const float* __restrict__ b_conv, float* __restrict__ nodes, long long N)
{
    const float wr = w_root[0], bc = b_conv[0];
    long long t  = (long long)blockIdx.x * blockDim.x + threadIdx.x;
    long long nv = N >> 2;
    if (t < nv) {
        v4f v = ((const v4f*)x)[t];
        v4f o;
        o.x = fmaf(v.x, wr, bc);
        o.y = fmaf(v.y, wr, bc);
        o.z = fmaf(v.z, wr, bc);
        o.w = fmaf(v.w, wr, bc);
        ((v4f*)nodes)[t] = o;
    }
    if (t == 0) {  // tail (empty for the given sizes)
        for (long long i = nv << 2; i < N; ++i) nodes[i] = fmaf(x[i], wr, bc);
    }
}

// ---------------------------------------------------------------------------
// Kernel 2: per-edge message + scatter-add.
// Streamed data (edge_index, edge_attr ~600MB) uses non-temporal loads so the
// 192MB L2 stays dedicated to x + nodes (~40MB resident -> gathers/atomics
// hit L2, not HBM).
// ---------------------------------------------------------------------------
__global__ __launch_bounds__(256) void edge_scatter_kernel(
    const float* __restrict__ x, const int* __restrict__ ei,
    const float* __restrict__ ea, const float* __restrict__ w_edge,
    const float* __restrict__ b_edge, float* __restrict__ nodes, long long E)
{
    const float we = w_edge[0], be = b_edge[0];
    long long t  = (long long)blockIdx.x * blockDim.x + threadIdx.x;
    long long nv = E >> 2;
    if (t < nv) {
        v4i s = __builtin_nontemporal_load((const v4i*)ei + t);       // src
        v4i d = __builtin_nontemporal_load((const v4i*)(ei + E) + t); // dst
        v4f a = __builtin_nontemporal_load((const v4f*)ea + t);
        atomicAdd(&nodes[d.x], x[s.x] * fmaf(a.x, we, be));
        atomicAdd(&nodes[d.y], x[s.y] * fmaf(a.y, we, be));
        atomicAdd(&nodes[d.z], x[s.z] * fmaf(a.z, we, be));
        atomicAdd(&nodes[d.w], x[s.w] * fmaf(a.w, we, be));
    }
    if (t == 0) {  // tail (empty for the given sizes)
        for (long long e = nv << 2; e < E; ++e)
            atomicAdd(&nodes[ei[E + e]], x[ei[e]] * fmaf(ea[e], we, be));
    }
}

// ---------------------------------------------------------------------------
// Kernel 3: per-graph head MLP (38->4->4->12, leaky_relu) + softmax.
// Each wave owns 16 graphs. Layer 1 = 10 x V_WMMA_F32_16X16X4_F32 (K=38 padded
// to 40); layers 2,3 = one WMMA each. 256 threads = 8 waves, G % 128 == 0 ->
// EXEC all-ones at every WMMA.
//
// All weight/bias padding is materialized ONCE per block into zero-padded LDS
// panels, so every B fragment / bias read in the hot path is an unconditional
// ds_load (no selects, no predicated loads, no exec-mask traffic).
//
// f32 WMMA fragment layouts (wave32):
//   A 16x4 : lane m=lane&15, hi=lane>>4 : {A[m][2hi], A[m][2hi+1]}
//   B 4x16 : lane n=lane&15, hi        : {B[2hi][n], B[2hi+1][n]}
//   C 16x16: vgpr r, lanes 0-15 -> (row r, col lane); lanes 16-31 -> (row r+8)
// ---------------------------------------------------------------------------
__global__ __launch_bounds__(256) void head_mlp_kernel(
    const float* __restrict__ nodes,
    const float* __restrict__ W1, const float* __restrict__ b1,
    const float* __restrict__ W2, const float* __restrict__ b2,
    const float* __restrict__ W3, const float* __restrict__ b3,
    float* __restrict__ out)
{
    __shared__ float panel1[16][40];   // panel1[n][k] = (n<4 && k<38) ? W1[k,n] : 0
    __shared__ float panel2[16][4];    // panel2[n][k] = (n<4)  ? W2[k,n] : 0
    __shared__ float panel3[16][4];    // panel3[n][k] = (n<12) ? W3[k,n] : 0
    __shared__ float b1p[16], b2p[16], b3p[16];   // zero-padded biases
    __shared__ float lds_h[8][16][16]; // per-wave staging tile

    const int tid  = threadIdx.x;
    const int lane = tid & 31;
    const int wave = tid >> 5;
    const int m    = lane & 15;        // A-row index / B,C column index
    const int hi   = lane >> 4;
    const int g0   = ((int)blockIdx.x * 8 + wave) * 16;  // first graph of tile

    // ---- one-time per-block: build zero-padded weight/bias panels in LDS ----
    #pragma unroll
    for (int i = tid; i < 16 * 40; i += 256) {
        const int n = i / 40, k = i % 40;
        panel1[n][k] = (n < 4 && k < NPG) ? W1[k * 4 + n] : 0.0f;
    }
    if (tid < 64) {
        const int n = tid >> 2, k = tid & 3;
        panel2[n][k] = (n < 4)  ? W2[k * 4 + n]  : 0.0f;
        panel3[n][k] = (n < 12) ? W3[k * 12 + n] : 0.0f;
    }
    if (tid < 16) {
        b1p[tid] = (tid < 4)  ? b1[tid] : 0.0f;
        b2p[tid] = (tid < 4)  ? b2[tid] : 0.0f;
        b3p[tid] = (tid < 12) ? b3[tid] : 0.0f;
    }
    __syncthreads();

    // -------- Layer 1: [16 x 38] @ [38 x 4], K padded to 40 --------
    v8f c1 = {};
    const float* arow = nodes + (long long)(g0 + m) * NPG;  // rows 8B-aligned
    #pragma unroll
    for (int k = 0; k < 9; ++k) {      // full K steps: all operands in range
        const int ka = 4 * k + 2 * hi;
        v2f a = *(const v2f*)(arow + ka);          // global_load_b64
        v2f b = *(const v2f*)&panel1[m][ka];       // ds_load_b64 (pre-masked)
        c1 = __builtin_amdgcn_wmma_f32_16x16x4_f32(
                 false, a, false, b, (short)0, c1, false, false);
    }
    {   // K tail (cols 36..39; 38,39 are padding -> clamp addr, select value)
        const int ka  = 36 + 2 * hi;
        const int ka0 = (ka     < NPG) ? ka     : 0;
        const int ka1 = (ka + 1 < NPG) ? ka + 1 : 0;
        float ax = arow[ka0], ay = arow[ka1];
        v2f a, b;
        a.x = (ka     < NPG) ? ax : 0.0f;
        a.y = (ka + 1 < NPG) ? ay : 0.0f;
        b = *(const v2f*)&panel1[m][ka];           // panel padded to 40 cols
        c1 = __builtin_amdgcn_wmma_f32_16x16x4_f32(
                 false, a, false, b, (short)0, c1, false, false);
    }
    {
        const float bias = b1p[m];
        #pragma unroll
        for (int r = 0; r < 8; ++r)
            lds_h[wave][r + 8 * hi][m] = leaky(c1[r] + bias);  // cols>=4 unused
    }
    __syncthreads();

    // -------- Layer 2: [16 x 4] @ [4 x 4], single WMMA --------
    v2f a2, bf2;
    a2.x = lds_h[wave][m][2 * hi];
    a2.y = lds_h[wave][m][2 * hi + 1];
    bf2  = *(const v2f*)&panel2[m][2 * hi];        // ds_load_b64 (pre-masked)
    v8f c2 = {};
    c2 = __builtin_amdgcn_wmma_f32_16x16x4_f32(
             false, a2, false, bf2, (short)0, c2, false, false);
    __syncthreads();
    {
        const float bias = b2p[m];
        #pragma unroll
        for (int r = 0; r < 8; ++r)
            lds_h[wave][r + 8 * hi][m] = leaky(c2[r] + bias);  // cols>=4 unused
    }
    __syncthreads();

    // -------- Layer 3: [16 x 4] @ [4 x 12], single WMMA --------
    v2f a3, bf3;
    a3.x = lds_h[wave][m][2 * hi];
    a3.y = lds_h[wave][m][2 * hi + 1];
    bf3  = *(const v2f*)&panel3[m][2 * hi];        // ds_load_b64 (pre-masked)
    v8f c3 = {};
    c3 = __builtin_amdgcn_wmma_f32_16x16x4_f32(
             false, a3, false, bf3, (short)0, c3, false, false);
    __syncthreads();
    {
        const float bias = b3p[m];
        #pragma unroll
        for (int r = 0; r < 8; ++r)
            lds_h[wave][r + 8 * hi][m] = leaky(c3[r] + bias); // cols>=12 unused
    }
    __syncthreads();

    // -------- Softmax over 12 classes: lane m (<16) owns graph g0+m --------
    if (lane < 16) {
        const float* hr = &lds_h[wave][m][0];
        float v[12];
        float mx = hr[0];
        #pragma unroll
        for (int j = 0; j < 12; ++j) { v[j] = hr[j]; mx = fmaxf(mx, v[j]); }
        float s = 0.0f;
        #pragma unroll
        for (int j = 0; j < 12; ++j) { v[j] = __expf(v[j] - mx); s += v[j]; }
        const float inv = __fdividef(1.0f, s);
        v4f* o = (v4f*)(out + (long long)(g0 + m) * 12);  // 48B-aligned
        v4f o0 = { v[0] * inv, v[1] * inv, v[2]  * inv, v[3]  * inv };
        v4f o1 = { v[4] * inv, v[5] * inv, v[6]  * inv, v[7]  * inv };
        v4f o2 = { v[8] * inv, v[9] * inv, v[10] * inv, v[11] * inv };
        o[0] = o0;
        o[1] = o1;
        o[2] = o2;
    }
}

// ---------------------------------------------------------------------------
extern "C" void kernel_launch(void* const* d_in, const int* in_sizes, int n_in,
                              void* d_out, int out_size, void* d_ws, size_t ws_size,
                              hipStream_t stream) {
    const float* x      = (const float*)d_in[0];
    const int*   ei     = (const int*)  d_in[1];   // [2, E] int32
    const float* ea     = (const float*)d_in[2];
    const float* w_edge = (const float*)d_in[3];
    const float* b_edge = (const float*)d_in[4];
    const float* w_root = (const float*)d_in[5];
    const float* b_conv = (const float*)d_in[6];
    const float* W1     = (const float*)d_in[7];
    const float* b1     = (const float*)d_in[8];
    const float* W2     = (const float*)d_in[9];
    const float* b2     = (const float*)d_in[10];
    const float* W3     = (const float*)d_in[11];
    const float* b3     = (const float*)d_in[12];
    float* out = (float*)d_out;

    const long long N = in_sizes[0];        // total nodes
    const long long E = in_sizes[2];        // total edges
    const int       G = (int)(N / NPG);     // graphs

    float* nodes = (float*)d_ws;            // N floats of scratch (~20 MB)

    {   // base term
        long long nv = N >> 2;
        unsigned blocks = (unsigned)((nv + 255) / 256);
        init_nodes_kernel<<<blocks, 256, 0, stream>>>(x, w_root, b_conv, nodes, N);
    }
    {   // edge messages + scatter add
        long long nv = E >> 2;
        unsigned blocks = (unsigned)((nv + 255) / 256);
        edge_scatter_kernel<<<blocks, 256, 0, stream>>>(x, ei, ea, w_edge, b_edge,
                                                        nodes, E);
    }
    {   // WMMA head MLP + softmax (16 graphs/wave, 8 waves/block)
        unsigned blocks = (unsigned)(G / 128);
        head_mlp_kernel<<<blocks, 256, 0, stream>>>(nodes, W1, b1, W2, b2, W3, b3,
                                                    out);
    }
}